// QFCModel_65481071398692
// MI455X (gfx1250) — compile-verified
//
#include <hip/hip_runtime.h>

typedef __attribute__((ext_vector_type(2))) float v2f;
typedef __attribute__((ext_vector_type(8))) float v8f;

#define NQ  4
#define DIM 16
#define PHI_STRIDE 20   // padded row stride (floats): 16B-aligned float4 stores,
                        // 8B-aligned b64 loads, bank-conflict-free

// ------------------------------------------------------------------
// Kernel 1: build the four real symmetric matrices
//   M[q][j][k] = sum_s Zq(s) * ( Re U[s,j] Re U[s,k] + Im U[s,j] Im U[s,k] )
// where U is the 16x16 unitary of 2 layers of (Rot on each qubit, CNOT ring).
// ws layout: float M[4][16][16]  (4 KB)
// ------------------------------------------------------------------
__global__ void qfc_precompute_M(const float* __restrict__ params,
                                 float* __restrict__ M) {
    __shared__ float cr[DIM][DIM];  // cr[j][s] = Re U[s, j]
    __shared__ float ci[DIM][DIM];
    const int tid = threadIdx.x;

    if (tid < DIM) {
        float re[DIM], im[DIM];
        #pragma unroll
        for (int s = 0; s < DIM; ++s) { re[s] = 0.f; im[s] = 0.f; }
        re[tid] = 1.f;  // basis state e_tid

        for (int l = 0; l < 2; ++l) {
            // Rot(phi, theta, omega) = RZ(omega) RY(theta) RZ(phi) on each qubit
            for (int q = 0; q < NQ; ++q) {
                const float phi = params[(l * NQ + q) * 3 + 0];
                const float th  = params[(l * NQ + q) * 3 + 1];
                const float om  = params[(l * NQ + q) * 3 + 2];
                float c, s, cP, sP, cM, sM;
                sincosf(th * 0.5f, &s, &c);
                sincosf((phi + om) * 0.5f, &sP, &cP);
                sincosf((phi - om) * 0.5f, &sM, &cM);
                // U00 = c*e^{-iP}, U01 = -s*e^{+iM}, U10 = s*e^{-iM}, U11 = c*e^{+iP}
                const float u00r =  c * cP, u00i = -c * sP;
                const float u01r = -s * cM, u01i = -s * sM;
                const float u10r =  s * cM, u10i = -s * sM;
                const float u11r =  c * cP, u11i =  c * sP;
                const int bit = 1 << q;
                #pragma unroll
                for (int base = 0; base < DIM; ++base) {
                    if (base & bit) continue;
                    const int i0 = base, i1 = base | bit;
                    const float a0r = re[i0], a0i = im[i0];
                    const float a1r = re[i1], a1i = im[i1];
                    re[i0] = u00r * a0r - u00i * a0i + u01r * a1r - u01i * a1i;
                    im[i0] = u00r * a0i + u00i * a0r + u01r * a1i + u01i * a1r;
                    re[i1] = u10r * a0r - u10i * a0i + u11r * a1r - u11i * a1i;
                    im[i1] = u10r * a0i + u10i * a0r + u11r * a1i + u11i * a1r;
                }
            }
            // CNOT ring: (0,1) (1,2) (2,3) (3,0): control=1 -> flip target (swap)
            #pragma unroll
            for (int e = 0; e < NQ; ++e) {
                const int cb = 1 << e, tb = 1 << ((e + 1) & 3);
                #pragma unroll
                for (int idx = 0; idx < DIM; ++idx) {
                    if ((idx & cb) && !(idx & tb)) {
                        const int j = idx | tb;
                        float tr = re[idx], ti = im[idx];
                        re[idx] = re[j]; im[idx] = im[j];
                        re[j] = tr;      im[j] = ti;
                    }
                }
            }
        }
        #pragma unroll
        for (int s = 0; s < DIM; ++s) { cr[tid][s] = re[s]; ci[tid][s] = im[s]; }
    }
    __syncthreads();

    const int j = tid >> 4, k = tid & 15;
    float m[NQ] = {0.f, 0.f, 0.f, 0.f};
    #pragma unroll
    for (int s = 0; s < DIM; ++s) {
        const float d = cr[j][s] * cr[k][s] + ci[j][s] * ci[k][s];
        #pragma unroll
        for (int q = 0; q < NQ; ++q)
            m[q] += ((s >> q) & 1) ? -d : d;
    }
    #pragma unroll
    for (int q = 0; q < NQ; ++q)
        M[q * 256 + j * 16 + k] = m[q];
}

// ------------------------------------------------------------------
// Kernel 2: per-row product state + WMMA quadratic forms.
// One thread per batch row; each wave32 owns 32 rows = two 16x16 tiles.
//   T_q = M_q x Phi^T   (A = M_q, B = Phi^T; 4 chained 16x16x4 f32 WMMAs)
//   ev[m,q] = sum_j Phi[m,j] * T_q[j,m]
// Column m of D lives on lanes {m, m+16}: the j-sum is an in-lane dot
// over the 8 accumulator VGPRs; half-sums combined through LDS.
// All lane-dependent selection (half) is folded into LDS ADDRESSES —
// no runtime indexing of register arrays, no divergence, no shuffles.
// ------------------------------------------------------------------
__global__ __launch_bounds__(256) void qfc_main(const float* __restrict__ x,
                                                const float* __restrict__ M,
                                                float* __restrict__ out) {
    __shared__ __align__(16) float phiLds[256 * PHI_STRIDE];   // 20 KB
    __shared__ __align__(16) float part[8 * 2 * 32 * 4];       // 8 KB partial evs

    const int tid  = threadIdx.x;
    const int lane = tid & 31;
    const int wave = tid >> 5;
    const int half = lane >> 4;      // 0: lanes 0-15, 1: lanes 16-31
    const int m16  = lane & 15;      // tile row/col index this lane serves
    const long row = (long)blockIdx.x * 256 + tid;

    // --- 1) RY angles: sum of 16 float4s (one row per thread) ---
    const float4* xr = (const float4*)(x + row * 64);
    float4 a = xr[0];
    #pragma unroll
    for (int i = 1; i < 16; ++i) {
        const float4 t = xr[i];
        a.x += t.x; a.y += t.y; a.z += t.z; a.w += t.w;
    }
    float c0, s0, c1, s1, c2, s2, c3, s3;
    sincosf(a.x * 0.5f, &s0, &c0);
    sincosf(a.y * 0.5f, &s1, &c1);
    sincosf(a.z * 0.5f, &s2, &c2);
    sincosf(a.w * 0.5f, &s3, &c3);

    // --- 2) product state phi[idx]: bit q of idx selects sin_q vs cos_q ---
    const float u0 = c1 * c0, u1 = c1 * s0, u2 = s1 * c0, u3 = s1 * s0;
    const float w0 = c3 * c2, w1 = c3 * s2, w2 = s3 * c2, w3 = s3 * s2;
    {
        float* myPhi = &phiLds[tid * PHI_STRIDE];
        float4 g0 = {u0 * w0, u1 * w0, u2 * w0, u3 * w0};
        float4 g1 = {u0 * w1, u1 * w1, u2 * w1, u3 * w1};
        float4 g2 = {u0 * w2, u1 * w2, u2 * w2, u3 * w2};
        float4 g3 = {u0 * w3, u1 * w3, u2 * w3, u3 * w3};
        *(float4*)(myPhi +  0) = g0;
        *(float4*)(myPhi +  4) = g1;
        *(float4*)(myPhi +  8) = g2;
        *(float4*)(myPhi + 12) = g3;
    }
    __syncthreads();

    // --- 3) A fragments (uniform): A[j, kk] = M_q[j = lane&15, 4c + kk]
    //        v0 = K={0|2}, v1 = K={1|3} per half-wave (ISA A-layout);
    //        'half' folded into the global address ---
    v2f Af[4][4];   // [q][c]
    {
        const float* mbase = M + m16 * 16 + 2 * half;
        #pragma unroll
        for (int q = 0; q < 4; ++q)
            #pragma unroll
            for (int c = 0; c < 4; ++c)
                Af[q][c] = *(const v2f*)(mbase + q * 256 + 4 * c);
    }

    const int waveBase = wave * 32;
    #pragma unroll
    for (int t = 0; t < 2; ++t) {
        const int rowBase = waveBase + t * 16;
        const float* src = &phiLds[(rowBase + m16) * PHI_STRIDE];

        // B fragments: B[kk, m=lane&15] = Phi[m, 4c + 2*half + {0,1}]
        // -> ds_load_b64 from runtime base (src + 2*half), imm offsets 4c
        const float* bsrc = src + 2 * half;
        v2f Bf[4];
        #pragma unroll
        for (int c = 0; c < 4; ++c)
            Bf[c] = *(const v2f*)(bsrc + 4 * c);

        // D-layout dot operand: Phi[m, r + 8*half], r = 0..7
        const float* dsrc = src + 8 * half;
        v2f pd[4];
        #pragma unroll
        for (int p = 0; p < 4; ++p)
            pd[p] = *(const v2f*)(dsrc + 2 * p);

        // T_q = M_q x Phi^T, then in-lane 8-element dot per q
        float pq[4];
        #pragma unroll
        for (int q = 0; q < 4; ++q) {
            v8f acc = {};
            #pragma unroll
            for (int c = 0; c < 4; ++c)
                acc = __builtin_amdgcn_wmma_f32_16x16x4_f32(
                    false, Af[q][c], false, Bf[c], (short)0, acc, false, false);
            float s = 0.f;
            #pragma unroll
            for (int p = 0; p < 4; ++p) {
                s = fmaf(acc[2 * p],     pd[p].x, s);
                s = fmaf(acc[2 * p + 1], pd[p].y, s);
            }
            pq[q] = s;
        }
        float4 ps = {pq[0], pq[1], pq[2], pq[3]};
        *(float4*)&part[((wave * 2 + t) * 32 + lane) * 4] = ps;
    }
    __syncthreads();

    // --- 4) combine the two half-sums for this thread's own row, store ---
    const int myT = (tid >> 4) & 1;
    const int base = ((wave * 2 + myT) * 32 + m16) * 4;
    const float4 p0 = *(const float4*)&part[base];
    const float4 p1 = *(const float4*)&part[base + 16 * 4];
    float4 ev;
    ev.x = p0.x + p1.x;
    ev.y = p0.y + p1.y;
    ev.z = p0.z + p1.z;
    ev.w = p0.w + p1.w;
    *(float4*)(out + row * 4) = ev;
}

extern "C" void kernel_launch(void* const* d_in, const int* in_sizes, int n_in,
                              void* d_out, int out_size, void* d_ws, size_t ws_size,
                              hipStream_t stream) {
    const float* x      = (const float*)d_in[0];   // (131072, 64) f32
    const float* params = (const float*)d_in[1];   // (2, 4, 3) f32
    float* out = (float*)d_out;                    // (131072, 4) f32
    float* M   = (float*)d_ws;                     // 4*16*16 floats

    qfc_precompute_M<<<1, 256, 0, stream>>>(params, M);

    const int rows   = in_sizes[0] / 64;
    const int blocks = rows / 256;                 // 131072 / 256 = 512
    qfc_main<<<blocks, 256, 0, stream>>>(x, M, out);
}